// LocalGrouper_56633438765134
// MI455X (gfx1250) — compile-verified
//
#include <hip/hip_runtime.h>
#include <hip/hip_bf16.h>
#include <math.h>

// Problem constants (match reference)
#define BB 4
#define NN 16384
#define CC 128
#define SS 1024
#define KK 32

#define CH   512          // points per chunk staged in LDS
#define ROWP (CH + 4)     // padded row stride for d2 tile buffer (bank-conflict break)

typedef float v2f __attribute__((ext_vector_type(2)));
typedef float v8f __attribute__((ext_vector_type(8)));

// ---- CDNA5 async global->LDS (ASYNCcnt-tracked) ----------------------------
__device__ __forceinline__ void async_load_f32_to_lds(const float* gaddr, float* lds_dst) {
    // generic pointer to __shared__ truncates to the 32-bit LDS offset (aperture rule)
    unsigned lds_off = (unsigned)(unsigned long long)lds_dst;
    asm volatile("global_load_async_to_lds_b32 %0, %1, off"
                 :: "v"(lds_off), "v"(gaddr) : "memory");
}
__device__ __forceinline__ void wait_async0() {
    asm volatile("s_wait_asynccnt 0" ::: "memory");
}

// ---- precompute ||x||^2 ----------------------------------------------------
__global__ void x2_kernel(const float* __restrict__ xyz, float* __restrict__ x2) {
    int i = blockIdx.x * blockDim.x + threadIdx.x;
    if (i < BB * NN) {
        const float* p = xyz + (size_t)i * 3;
        x2[i] = p[0] * p[0] + p[1] * p[1] + p[2] * p[2];
    }
}

// ---- gather centers (output #1) and ||c||^2 --------------------------------
__global__ void center_kernel(const float* __restrict__ xyz, const int* __restrict__ idx,
                              float* __restrict__ center_out, float* __restrict__ c2) {
    int i = blockIdx.x * blockDim.x + threadIdx.x;
    if (i < BB * SS) {
        int b = i / SS, s = i % SS;
        int j = idx[s];
        const float* p = xyz + ((size_t)b * NN + j) * 3;
        float x = p[0], y = p[1], z = p[2];
        float* o = center_out + (size_t)i * 3;
        o[0] = x; o[1] = y; o[2] = z;
        c2[i] = x * x + y * y + z * z;
    }
}

// ---- kNN: WMMA f32 16x16x4 distance tiles + wave32 bitonic top-32 ----------
// Grid: BB*(SS/16) blocks of 512 threads (16 waves). Wave w: compute 2 tiles,
// then own center row w for selection.
__launch_bounds__(512)
__global__ void knn_kernel(const float* __restrict__ xyz, const float* __restrict__ center,
                           const float* __restrict__ x2g, const float* __restrict__ c2g,
                           int* __restrict__ knn) {
    __shared__ v2f   lds_xy[CH];        // {x, y} per point
    __shared__ v2f   lds_zw[CH];        // {z, 0} per point
    __shared__ float lds_x2[CH];
    __shared__ float lds_d2[16 * ROWP];

    const int tid  = threadIdx.x;
    const int lane = tid & 31;
    const int w    = tid >> 5;
    const int b    = blockIdx.x / (SS / 16);
    const int s0   = (blockIdx.x % (SS / 16)) * 16;
    const int hi   = (lane >= 16);
    const int am   = lane & 15;

    // A operand: centers 16x4 f32 (K=3 padded). Lanes 0-15: {K0,K1}=(x,y);
    // lanes 16-31: {K2,K3}=(z,0). Identical in every wave. Branchless build.
    v2f a;
    {
        const float* cp = center + ((size_t)b * SS + s0 + am) * 3;
        a[0] = cp[hi ? 2 : 0];
        float cy = cp[1];
        a[1] = hi ? 0.0f : cy;
    }
    // c2 for the 8 center rows this lane's D-fragment covers (M = r + 8*hi)
    float c2v[8];
#pragma unroll
    for (int r = 0; r < 8; ++r)
        c2v[r] = c2g[b * SS + s0 + r + (hi ? 8 : 0)];

    const float* xyz_b = xyz + (size_t)b * NN * 3;
    const float* x2_b  = x2g + (size_t)b * NN;

    // running top-32 (one entry per lane, ascending by (dist, idx) across lanes)
    float top_v = INFINITY;
    int   top_i = 0x7fffffff;

    for (int n0 = 0; n0 < NN; n0 += CH) {
        // ---- stage chunk in LDS (512 threads == CH points) ----
        {
            const float* p = xyz_b + (size_t)(n0 + tid) * 3;
            float x = p[0], y = p[1], z = p[2];
            v2f xy; xy[0] = x; xy[1] = y;
            v2f zw; zw[0] = z; zw[1] = 0.0f;
            lds_xy[tid] = xy;
            lds_zw[tid] = zw;
            async_load_f32_to_lds(x2_b + n0 + tid, &lds_x2[tid]);
            if (n0 + CH < NN)
                __builtin_prefetch(xyz_b + (size_t)(n0 + CH + tid) * 3, 0, 0);
            wait_async0();
        }
        __syncthreads();

        // ---- compute phase: wave w -> tiles j = w, w+16 ----
        // Branchless B operand: lane-selected pointer -> single ds_load_b64.
        const v2f* bsrc = hi ? lds_zw : lds_xy;
#pragma unroll
        for (int jj = 0; jj < 2; ++jj) {
            const int j   = w + jj * 16;
            const int col = j * 16 + am;
            v2f bop = bsrc[col];   // lanes 0-15: {x,y}; lanes 16-31: {z,0}
            v8f acc = {};
            acc = __builtin_amdgcn_wmma_f32_16x16x4_f32(
                false, a, false, bop, (short)0, acc, false, false);
            const float xx = lds_x2[col];
#pragma unroll
            for (int r = 0; r < 8; ++r) {
                const int m = r + (hi ? 8 : 0);
                lds_d2[m * ROWP + col] = c2v[r] + xx - 2.0f * acc[r];
            }
        }
        __syncthreads();

        // ---- select phase: wave w scans row m = w ----
        {
            const float* row = &lds_d2[w * ROWP];
            for (int t0 = 0; t0 < CH; t0 += 32) {
                float v  = row[t0 + lane];
                int   gi = n0 + t0 + lane;
                const float tv = __shfl(top_v, 31);   // current worst kept
                const int   ti = __shfl(top_i, 31);
                const bool cand = (v < tv) || (v == tv && gi < ti);
                if (__any(cand)) {                    // wave-uniform branch
                    // bitonic sort the 32 new candidates ascending by (v, gi)
#pragma unroll
                    for (int kk = 2; kk <= 32; kk <<= 1) {
#pragma unroll
                        for (int jm = kk >> 1; jm >= 1; jm >>= 1) {
                            const float ov = __shfl_xor(v, jm);
                            const int   oi = __shfl_xor(gi, jm);
                            const bool up      = ((lane & kk) == 0);
                            const bool lower   = ((lane & jm) == 0);
                            const bool takeMin = (lower == up);
                            const bool less    = (ov < v) || (ov == v && oi < gi);
                            if (takeMin == less) { v = ov; gi = oi; }
                        }
                    }
                    // Batcher merge: min(sorted_top[i], sorted_new[31-i]) = 32 smallest
                    const float rv = __shfl(v, 31 - lane);
                    const int   ri = __shfl(gi, 31 - lane);
                    const bool less2 = (rv < top_v) || (rv == top_v && ri < top_i);
                    if (less2) { top_v = rv; top_i = ri; }
                    // cleanup: bitonic sequence -> ascending
#pragma unroll
                    for (int jm = 16; jm >= 1; jm >>= 1) {
                        const float ov = __shfl_xor(top_v, jm);
                        const int   oi = __shfl_xor(top_i, jm);
                        const bool lower = ((lane & jm) == 0);
                        const bool less3 = (ov < top_v) || (ov == top_v && oi < top_i);
                        if (lower == less3) { top_v = ov; top_i = oi; }
                    }
                }
            }
        }
        __syncthreads();
    }

    knn[((size_t)b * SS + s0 + w) * KK + lane] = top_i;
}

// ---- build new_points (output #2) ------------------------------------------
__global__ void gather_kernel(const float* __restrict__ xyz, const float* __restrict__ points,
                              const float* __restrict__ center, const int* __restrict__ knn,
                              float* __restrict__ out_np) {
    long long t = (long long)blockIdx.x * blockDim.x + threadIdx.x;
    const long long total = (long long)BB * SS * KK * (CC + 3);
    if (t >= total) return;
    const int c = (int)(t % (CC + 3));
    long long row = t / (CC + 3);
    const int k = (int)(row % KK);
    long long sr = row / KK;
    const int s = (int)(sr % SS);
    const int b = (int)(sr / SS);
    const int j = knn[((size_t)b * SS + s) * KK + k];
    float val;
    if (c < 3)
        val = xyz[((size_t)b * NN + j) * 3 + c] - center[((size_t)b * SS + s) * 3 + c];
    else
        val = points[((size_t)b * NN + j) * CC + (c - 3)];
    out_np[t] = val;
}

extern "C" void kernel_launch(void* const* d_in, const int* in_sizes, int n_in,
                              void* d_out, int out_size, void* d_ws, size_t ws_size,
                              hipStream_t stream) {
    const float* xyz    = (const float*)d_in[0];
    const float* points = (const float*)d_in[1];
    const int*   idx    = (const int*)d_in[2];

    float* out        = (float*)d_out;
    float* center_out = out;                        // B*S*3 floats
    float* out_np     = out + (size_t)BB * SS * 3;  // B*S*K*(C+3) floats

    // workspace: x2 (B*N f32) | c2 (B*S f32) | knn (B*S*K i32)  ~784 KB
    float* ws_x2  = (float*)d_ws;
    float* ws_c2  = ws_x2 + (size_t)BB * NN;
    int*   ws_knn = (int*)(ws_c2 + (size_t)BB * SS);

    x2_kernel<<<(BB * NN + 255) / 256, 256, 0, stream>>>(xyz, ws_x2);
    center_kernel<<<(BB * SS + 255) / 256, 256, 0, stream>>>(xyz, idx, center_out, ws_c2);
    knn_kernel<<<BB * (SS / 16), 512, 0, stream>>>(xyz, center_out, ws_x2, ws_c2, ws_knn);

    const long long total = (long long)BB * SS * KK * (CC + 3);
    gather_kernel<<<(unsigned)((total + 255) / 256), 256, 0, stream>>>(
        xyz, points, center_out, ws_knn, out_np);
}